// PatchIoULoss_50637664420512
// MI455X (gfx1250) — compile-verified
//
#include <hip/hip_runtime.h>

typedef float v2f __attribute__((ext_vector_type(2)));
typedef float v8f __attribute__((ext_vector_type(8)));

#define IMGS 16
#define HW   1024
#define BLK  32            // 32x32 block sums (64x64 window = 2x2 blocks)
#define GB   (HW / BLK)    // 32 blocks per dimension
#define NBLK (GB * GB)     // 1024 blocks per image
#define WOUT 31            // (1024-64)/32 + 1 windows per dimension
#define NWIN (WOUT * WOUT) // 961 windows per image
#define EPSV 1e-7f

// ds_swizzle_b32 XOR folds (group-of-32 mode: offset[14:10]=xor, [4:0]=0x1F).
// One DS op + one add per fold; no per-lane index math.
#define SWZ_XOR16 0x401F
#define SWZ_XOR8  0x201F
#define SWZ_XOR4  0x101F
#define SWZ_XOR2  0x081F
#define SWZ_XOR1  0x041F

template <int IMM>
__device__ __forceinline__ float fold_add(float x) {
    int y = __builtin_amdgcn_ds_swizzle(__float_as_int(x), IMM);
    return x + __int_as_float(y);
}

// ---------------------------------------------------------------------------
// Kernel 1: streaming pass. One 256-thread WG per 32x32 tile (16384 WGs).
// Thread t: one float4 per input at (row=t/8, cols=4*(t%8)) -> 128B-coalesced.
// All threads write 3 partials to LDS; wave 0 reduces all 256 partials per
// quantity with 4 chained V_WMMA_F32_16X16X4_F32 (ones-A, C accumulate):
//   chunk j: B[k][n] = part[64j + 16k + n]  =>  D[.][n] = sum_j,k part[...]
// leaving total[n%16] in every lane, then 4 ds_swizzle XOR folds finish 16->1.
// ---------------------------------------------------------------------------
__global__ __launch_bounds__(256)
void block_sums_kernel(const float* __restrict__ pred,
                       const float* __restrict__ targ,
                       float* __restrict__ bs /* [3][IMGS*NBLK] */) {
    const int wg  = blockIdx.x;        // 0 .. IMGS*NBLK-1
    const int img = wg / NBLK;
    const int blk = wg % NBLK;
    const int by  = blk / GB;
    const int bx  = blk % GB;
    const int t   = threadIdx.x;

    const int row = by * BLK + (t >> 3);
    const int col = bx * BLK + ((t & 7) << 2);
    const size_t base = ((size_t)img * HW + row) * HW + col;

    const float4 p = *reinterpret_cast<const float4*>(pred + base);
    const float4 q = *reinterpret_cast<const float4*>(targ + base);

    const float sp = (p.x + p.y) + (p.z + p.w);
    const float st = (q.x + q.y) + (q.z + q.w);
    const float si = fmaf(p.x, q.x, fmaf(p.y, q.y, fmaf(p.z, q.z, p.w * q.w)));

    __shared__ float part[3][256];
    part[0][t] = sp;
    part[1][t] = st;
    part[2][t] = si;
    __syncthreads();

    if (t < 32) {                      // wave 0 only: EXEC all-1s inside
        // B-matrix gather index: lanes 0-15 -> K0/K1 rows, lanes 16-31 -> K2/K3.
        const int bidx = (t & 15) + ((t & 16) << 1);   // +0 or +32
        v2f a; a[0] = 1.0f; a[1] = 1.0f;               // A = ones(16x4)

        float out0 = 0.f, out1 = 0.f, out2 = 0.f;
#pragma unroll
        for (int qn = 0; qn < 3; ++qn) {
            v8f c = {};
#pragma unroll
            for (int j = 0; j < 4; ++j) {
                v2f b;
                b[0] = part[qn][bidx + 64 * j];
                b[1] = part[qn][bidx + 64 * j + 16];
                c = __builtin_amdgcn_wmma_f32_16x16x4_f32(false, a, false, b,
                                                          (short)0, c, false, false);
            }
            float s = c[0];                            // total[lane % 16]
            s = fold_add<SWZ_XOR8>(s);
            s = fold_add<SWZ_XOR4>(s);
            s = fold_add<SWZ_XOR2>(s);
            s = fold_add<SWZ_XOR1>(s);
            if (qn == 0) out0 = s; else if (qn == 1) out1 = s; else out2 = s;
        }

        if (t == 0) {
            const int idx = img * NBLK + blk;
            bs[idx]                   = out0;  // plane 0: sum(pred)
            bs[IMGS * NBLK + idx]     = out1;  // plane 1: sum(target)
            bs[2 * IMGS * NBLK + idx] = out2;  // plane 2: sum(pred*target)
        }
    }
}

// ---------------------------------------------------------------------------
// Kernel 2: one WG per image; each 64x64 window = 2x2 adjacent block sums.
// ---------------------------------------------------------------------------
__global__ __launch_bounds__(256)
void window_loss_kernel(const float* __restrict__ bs,
                        float* __restrict__ partial /* [IMGS] */) {
    const int img = blockIdx.x;
    const int t   = threadIdx.x;
    const float* P = bs + (size_t)img * NBLK;
    const float* T = bs + (size_t)IMGS * NBLK + (size_t)img * NBLK;
    const float* I = bs + (size_t)2 * IMGS * NBLK + (size_t)img * NBLK;

    float acc = 0.0f;
    for (int w = t; w < NWIN; w += 256) {
        const int wy = w / WOUT;
        const int wx = w - wy * WOUT;
        const int i00 = wy * GB + wx;
        const float pi = (P[i00] + P[i00 + 1]) + (P[i00 + GB] + P[i00 + GB + 1]);
        const float ti = (T[i00] + T[i00 + 1]) + (T[i00 + GB] + T[i00 + GB + 1]);
        const float ii = (I[i00] + I[i00 + 1]) + (I[i00 + GB] + I[i00 + GB + 1]);
        const float un = pi + ti - ii;
        acc += 1.0f - (ii + EPSV) / (un + EPSV);
    }

    acc = fold_add<SWZ_XOR16>(acc);
    acc = fold_add<SWZ_XOR8>(acc);
    acc = fold_add<SWZ_XOR4>(acc);
    acc = fold_add<SWZ_XOR2>(acc);
    acc = fold_add<SWZ_XOR1>(acc);

    __shared__ float red[8];
    const int lane = t & 31;
    const int wave = t >> 5;
    if (lane == 0) red[wave] = acc;
    __syncthreads();
    if (t == 0) {
        float s = 0.f;
#pragma unroll
        for (int i = 0; i < 8; ++i) s += red[i];
        partial[img] = s;
    }
}

// ---------------------------------------------------------------------------
// Kernel 3: fold 16 per-image partials into the scalar mean.
// ---------------------------------------------------------------------------
__global__ void finalize_kernel(const float* __restrict__ partial,
                                float* __restrict__ out) {
    float v = (threadIdx.x < IMGS) ? partial[threadIdx.x] : 0.0f;
    v = fold_add<SWZ_XOR8>(v);
    v = fold_add<SWZ_XOR4>(v);
    v = fold_add<SWZ_XOR2>(v);
    v = fold_add<SWZ_XOR1>(v);
    if (threadIdx.x == 0) out[0] = v / (float)(IMGS * NWIN);
}

extern "C" void kernel_launch(void* const* d_in, const int* in_sizes, int n_in,
                              void* d_out, int out_size, void* d_ws, size_t ws_size,
                              hipStream_t stream) {
    (void)in_sizes; (void)n_in; (void)out_size; (void)ws_size;
    const float* pred = (const float*)d_in[0];
    const float* targ = (const float*)d_in[1];

    float* bs      = (float*)d_ws;             // 3 * 16 * 1024 floats (~192 KB)
    float* partial = bs + 3 * IMGS * NBLK;     // +16 floats

    block_sums_kernel<<<IMGS * NBLK, 256, 0, stream>>>(pred, targ, bs);
    window_loss_kernel<<<IMGS, 256, 0, stream>>>(bs, partial);
    finalize_kernel<<<1, 32, 0, stream>>>(partial, (float*)d_out);
}